// Net_48180943127420
// MI455X (gfx1250) — compile-verified
//
#include <hip/hip_runtime.h>

// GCN conv (scatter-add over 80M edges) + per-graph FC on MI455X (gfx1250).
//
// HBM traffic budget: 640MB (col pass) + 1.28GB (row+col pass) ~= 1.92GB
// -> ~82us at 23.3 TB/s. All node-sized arrays (<=60MB scratch) fit in the
// 192MB L2, so gathers/atomics stay in L2. Edge stream uses non-temporal
// loads to avoid evicting the hot set. Final [G,50]x[50,2] FC uses
// V_WMMA_F32_16X16X4_F32 (fp32-exact WMMA) with a branch-free inner loop.

typedef __attribute__((ext_vector_type(2))) float      v2f;
typedef __attribute__((ext_vector_type(8))) float      v8f;
typedef __attribute__((ext_vector_type(2))) long long  v2ll;

#define NPG 50   // nodes per graph (reference: NODES_PER_GRAPH)

// ---------------- pass 0: zero the degree buffer ----------------
__global__ void k_zero_u32(unsigned* __restrict__ p, int n) {
    int i = blockIdx.x * blockDim.x + threadIdx.x;
    if (i < n) p[i] = 0u;
}

// ---------------- pass 1: in-degree (excl. self loop) ----------------
// 4 edges/thread, NT b128 loads of the int64 col stream.
__global__ __launch_bounds__(256) void k_degree(const long long* __restrict__ col,
                                                unsigned* __restrict__ deg, int E) {
    int base = (blockIdx.x * blockDim.x + threadIdx.x) * 4;
    if (base + 3 < E) {
        v2ll c01 = __builtin_nontemporal_load((const v2ll*)(col + base));
        v2ll c23 = __builtin_nontemporal_load((const v2ll*)(col + base + 2));
        atomicAdd(&deg[(int)c01.x], 1u);
        atomicAdd(&deg[(int)c01.y], 1u);
        atomicAdd(&deg[(int)c23.x], 1u);
        atomicAdd(&deg[(int)c23.y], 1u);
    } else {
        for (int e = base; e < E; ++e)
            atomicAdd(&deg[(int)col[e]], 1u);
    }
}

// ---------------- pass 2: dis = rsqrt(deg+1), s = x*w*dis; re-zero deg->agg --
__global__ void k_prep(const float* __restrict__ x, const float* __restrict__ conv_w,
                       unsigned* __restrict__ deg_agg, float* __restrict__ dis,
                       float* __restrict__ s, int n) {
    int i = blockIdx.x * blockDim.x + threadIdx.x;
    if (i >= n) return;
    float w  = conv_w[0];
    float d  = (float)(deg_agg[i] + 1u);   // + self loop, always >= 1
    float di = rsqrtf(d);
    dis[i] = di;
    s[i]   = x[i] * w * di;                // pre-scaled source value
    ((float*)deg_agg)[i] = 0.0f;           // buffer becomes float agg
}

// ---------------- pass 3: scatter messages: agg[c] += s[r]*dis[c] -----------
__global__ __launch_bounds__(256) void k_scatter(const long long* __restrict__ row,
                                                 const long long* __restrict__ col,
                                                 const float* __restrict__ s,
                                                 const float* __restrict__ dis,
                                                 float* __restrict__ agg, int E) {
    int base = (blockIdx.x * blockDim.x + threadIdx.x) * 4;
    if (base + 3 < E) {
        v2ll r01 = __builtin_nontemporal_load((const v2ll*)(row + base));
        v2ll r23 = __builtin_nontemporal_load((const v2ll*)(row + base + 2));
        v2ll c01 = __builtin_nontemporal_load((const v2ll*)(col + base));
        v2ll c23 = __builtin_nontemporal_load((const v2ll*)(col + base + 2));
        int r0 = (int)r01.x, r1 = (int)r01.y, r2 = (int)r23.x, r3 = (int)r23.y;
        int c0 = (int)c01.x, c1 = (int)c01.y, c2 = (int)c23.x, c3 = (int)c23.y;
        atomicAdd(&agg[c0], s[r0] * dis[c0]);
        atomicAdd(&agg[c1], s[r1] * dis[c1]);
        atomicAdd(&agg[c2], s[r2] * dis[c2]);
        atomicAdd(&agg[c3], s[r3] * dis[c3]);
    } else {
        for (int e = base; e < E; ++e) {
            int r = (int)row[e], c = (int)col[e];
            atomicAdd(&agg[c], s[r] * dis[c]);
        }
    }
}

// ---------------- pass 4: h = agg + self_loop + b  (h overwrites s) ---------
__global__ void k_finalize(const float* __restrict__ agg, const float* __restrict__ dis,
                           const float* __restrict__ conv_b, float* __restrict__ s_h, int n) {
    int i = blockIdx.x * blockDim.x + threadIdx.x;
    if (i >= n) return;
    // self-loop message: xw[i]*dis[i]^2 == s[i]*dis[i]
    s_h[i] = agg[i] + s_h[i] * dis[i] + conv_b[0];
}

// ---------------- pass 5: per-graph FC via V_WMMA_F32_16X16X4_F32 -----------
// One wave per 16 graphs. A = h tile [16 x 52] (K padded 50->52), B[k][n] =
// fc_w[n][k] for n<2 else 0, D[m][n] accumulated over 13 K=4 WMMAs.
// ISA layouts (CDNA5 ISA):
//   A 16x4 f32 : lanes 0-15 -> K={0,1} in v0,v1 ; lanes 16-31 -> K={2,3}
//   B  4x16 f32: v0 = rows K=0(lo)/K=2(hi), v1 = rows K=1(lo)/K=3(hi)
//   C/D 16x16  : vr -> M=r (lanes 0-15), M=r+8 (lanes 16-31), N = lane&15
//
// Branch-free: only B is masked (via float multiply); A/B addresses are
// clamped into valid memory so every load is unconditional and EXEC stays
// all-1s -- no saveexec/branch per load, loads batch ahead of the WMMA chain.
__global__ __launch_bounds__(32) void k_fc_wmma(const float* __restrict__ h,
                                                const float* __restrict__ fcw,
                                                const float* __restrict__ fcb,
                                                float* __restrict__ out, int G) {
    int lane  = threadIdx.x;          // 0..31, one wave per block
    int gbase = blockIdx.x * 16;
    int mrow  = lane & 15;
    int hi    = lane >> 4;            // 0: lanes 0-15, 1: lanes 16-31
    int gm    = gbase + mrow;
    if (gm > G - 1) gm = G - 1;       // clamp tail reads into valid memory
    const float* arow = h + (long long)gm * NPG;

    int   brow  = (mrow < 2) ? mrow : 0;         // clamped B row
    float bmrow = (mrow < 2) ? 1.0f : 0.0f;      // N-mask for B

    v8f acc = {};
#pragma unroll
    for (int kc = 0; kc < 13; ++kc) {
        int k0 = kc * 4 + hi * 2;                // this lane's two K indices
        int ka = (k0 <= NPG - 2) ? k0 : (NPG - 2); // clamp pair into range (even)
        // A: one aligned b64 load, unmasked (zeros come from masked B)
        v2f a = *(const v2f*)(arow + ka);
        // B: clamped b32 loads, zeroed by multiply for invalid k or n>=2
        float bk0 = (k0     < NPG) ? bmrow : 0.0f;
        float bk1 = (k0 + 1 < NPG) ? bmrow : 0.0f;
        v2f b;
        b.x = fcw[brow * NPG + ka]     * bk0;
        b.y = fcw[brow * NPG + ka + 1] * bk1;
        acc = __builtin_amdgcn_wmma_f32_16x16x4_f32(
            /*neg_a=*/false, a, /*neg_b=*/false, b,
            /*c_mod=*/(short)0, acc, /*reuse_a=*/false, /*reuse_b=*/false);
    }

    int nn = mrow;                    // N = lane & 15
    if (nn < 2) {
        float bias = fcb[nn];
        int mofs = hi * 8;
#pragma unroll
        for (int r = 0; r < 8; ++r) {
            int g = gbase + mofs + r;
            if (g < G) out[g * 2 + nn] = acc[r] + bias;
        }
    }
}

extern "C" void kernel_launch(void* const* d_in, const int* in_sizes, int n_in,
                              void* d_out, int out_size, void* d_ws, size_t ws_size,
                              hipStream_t stream) {
    const float*     x      = (const float*)d_in[0];
    const long long* eidx   = (const long long*)d_in[1];   // int64 [2, E]
    const float*     conv_w = (const float*)d_in[2];
    const float*     conv_b = (const float*)d_in[3];
    const float*     fc_w   = (const float*)d_in[4];       // [2, 50]
    const float*     fc_b   = (const float*)d_in[5];       // [2]
    float*           out    = (float*)d_out;               // [G, 2]

    const int N = in_sizes[0];            // 5,000,000
    const int E = in_sizes[1] / 2;        // 80,000,000
    const int G = N / NPG;                // 100,000

    const long long* row = eidx;          // sources
    const long long* col = eidx + E;      // targets

    // workspace: [deg(u32)->agg(f32) | dis(f32) | s(f32)->h(f32)] = 3*N*4 bytes
    unsigned* deg_agg = (unsigned*)d_ws;
    float*    dis     = (float*)((char*)d_ws + (size_t)N * 4);
    float*    s_h     = (float*)((char*)d_ws + (size_t)N * 8);

    const int TB = 256;
    const int nodeBlocks = (N + TB - 1) / TB;
    const int edgeThreads = (E + 3) / 4;
    const int edgeBlocks = (edgeThreads + TB - 1) / TB;

    k_zero_u32<<<nodeBlocks, TB, 0, stream>>>(deg_agg, N);
    k_degree  <<<edgeBlocks, TB, 0, stream>>>(col, deg_agg, E);
    k_prep    <<<nodeBlocks, TB, 0, stream>>>(x, conv_w, deg_agg, dis, s_h, N);
    k_scatter <<<edgeBlocks, TB, 0, stream>>>(row, col, s_h, dis, (float*)deg_agg, E);
    k_finalize<<<nodeBlocks, TB, 0, stream>>>((const float*)deg_agg, dis, conv_b, s_h, N);

    const int tiles = (G + 15) / 16;      // 6250 waves, one per 16 graphs
    k_fc_wmma <<<tiles, 32, 0, stream>>>(s_h, fc_w, fc_b, out, G);
}